// SimpleAttention_57019985822208
// MI455X (gfx1250) — compile-verified
//
#include <hip/hip_runtime.h>

// ---------------------------------------------------------------------------
// Problem constants (from reference): BSZ=8, SEQ=4096, H=32, D=128, R=1536
// ---------------------------------------------------------------------------
#define BSZ 8
#define SEQ 4096
#define NH  32
#define HD  128
#define RR  1536
#define NCOL (NH * HD)   // 4096

typedef __attribute__((ext_vector_type(2))) float v2f;
typedef __attribute__((ext_vector_type(4))) float v4f;
typedef __attribute__((ext_vector_type(8))) float v8f;

// ---------------------------------------------------------------------------
// Kernel 1: q projection  q[b,n] = sum_r hs[b,r] * w_q[n,r] + b_q[n]
// GEMM M=8 (padded to 16), N=4096, K=1536 using V_WMMA_F32_16X16X4_F32.
// One wave per 16-wide N tile. A rows 8..15 mirror rows 0..7 (finite,
// discarded at store). Fragment layouts per CDNA5 ISA:
//   A 16x4 f32 : vgpr r, lanes 0-15 -> K=r, lanes 16-31 -> K=r+2, M=lane%16
//   B 4x16 f32 : vgpr r, lanes 0-15 -> K=r, lanes 16-31 -> K=r+2, N=lane%16
//   C 16x16    : vgpr r, lanes 0-15 -> M=r, N=lane; lanes 16-31 -> M=r+8
// ---------------------------------------------------------------------------
__global__ __launch_bounds__(32) void qproj_wmma_kernel(
    const float* __restrict__ hs,   // [8, 1536]
    const float* __restrict__ wq,   // [4096, 1536]
    const float* __restrict__ bq,   // [4096]
    float* __restrict__ qout)       // [8, 4096]
{
    const int lane  = threadIdx.x;        // 0..31
    const int ntile = blockIdx.x;         // 0..255
    const int lm    = lane & 15;
    const int hi    = lane >> 4;          // 0 or 1 -> K offset +2
    const int n     = ntile * 16 + lm;

    const float* arow = hs + (size_t)(lm & 7) * RR;   // pad rows mirror real ones
    const float* brow = wq + (size_t)n * RR;

    v8f c = {};
    #pragma unroll 4
    for (int kk = 0; kk < RR / 4; ++kk) {
        const int kb = 4 * kk + 2 * hi;
        v2f a = *(const v2f*)(arow + kb);   // {A[m][kb], A[m][kb+1]}
        v2f b = *(const v2f*)(brow + kb);   // {B[kb][n], B[kb+1][n]}
        c = __builtin_amdgcn_wmma_f32_16x16x4_f32(
                /*neg_a=*/false, a, /*neg_b=*/false, b,
                /*c_mod=*/(short)0, c, /*reuse_a=*/false, /*reuse_b=*/false);
    }

    if (lane < 16) {                       // lanes 0-15 hold M=0..7 in vgprs 0..7
        const float bias = bq[n];
        #pragma unroll
        for (int r = 0; r < 8; ++r)
            qout[(size_t)r * NCOL + n] = c[r] + bias;
    }
}

// ---------------------------------------------------------------------------
// Kernel 2: flash-style attention for one (b,h) per workgroup (8 waves).
// Single pass over S: K and V each read exactly once (HBM-bound optimum,
// ~1 GiB / 23.3 TB/s ~= 46 us floor). Scores via WMMA with the q-vector
// replicated across all 16 A rows, so every D row equals the score row;
// lane l reads the score for s0+(l%16) from C vgpr0. Online softmax per
// wave; V accumulated with coalesced non-temporal v4f loads (lane l owns
// d = 4l..4l+3). K/V are zero-reuse -> NT loads keep L2 clean; speculative
// global_prefetch_b8 runs ahead on the 16KB-strided key rows.
// ---------------------------------------------------------------------------
__global__ __launch_bounds__(256) void attn_flash_wmma_kernel(
    const float* __restrict__ qall,   // [8, 4096]  (= [b, h*128 + d])
    const float* __restrict__ key,    // [8, 4096, 32, 128]
    const float* __restrict__ value,  // [8, 4096, 32, 128]
    float* __restrict__ out)          // [8, 4096]
{
    const int bh   = blockIdx.x;          // 0..255
    const int b    = bh >> 5;
    const int h    = bh & 31;
    const int tid  = threadIdx.x;
    const int wave = tid >> 5;            // 0..7
    const int lane = tid & 31;
    const int lm   = lane & 15;
    const int hi   = lane >> 4;

    // --- preload q A-fragments (replicated rows): aq[kk] covers d=4kk..4kk+3
    const float* qrow = qall + (size_t)b * NCOL + (size_t)h * HD;
    v2f aq[32];
    #pragma unroll
    for (int kk = 0; kk < 32; ++kk)
        aq[kk] = *(const v2f*)(qrow + 4 * kk + 2 * hi);

    const size_t rs = (size_t)NH * HD;    // stride between consecutive s: 4096 floats
    const float* kbase = key   + ((size_t)b * SEQ * NH + h) * HD;
    const float* vbase = value + ((size_t)b * SEQ * NH + h) * HD;

    float m_run = -3.0e38f;
    float l_run = 0.0f;
    v4f   o     = {0.f, 0.f, 0.f, 0.f};

    for (int t = wave; t < SEQ / 16; t += 8) {
        const int s0 = t * 16;
        const float* krow = kbase + (size_t)(s0 + lm) * rs;  // this lane's key row

        // --- speculative prefetch of this wave's NEXT tile (s0+128):
        //     2 prefetches/lane cover the 4x128B lines of each 512B row.
        {
            const char* kn = (const char*)(krow + 128 * rs) + hi * 256;
            const char* vn = (const char*)(vbase + (size_t)(s0 + 128 + lm) * rs) + hi * 256;
            __builtin_prefetch(kn, 0, 0);
            __builtin_prefetch(kn + 128, 0, 0);
            __builtin_prefetch(vn, 0, 0);
            __builtin_prefetch(vn + 128, 0, 0);
        }

        // --- scores for 16 rows via 32 x wmma_f32_16x16x4 over d=0..127
        v8f c = {};
        #pragma unroll 8
        for (int kk = 0; kk < 32; ++kk) {
            v2f kb2 = __builtin_nontemporal_load((const v2f*)(krow + 4 * kk + 2 * hi));
            c = __builtin_amdgcn_wmma_f32_16x16x4_f32(
                    false, aq[kk], false, kb2, (short)0, c, false, false);
        }
        float score = c[0];               // score for s = s0 + lm (lane pairs duplicate)

        // --- online softmax update (wave32 reductions; columns counted twice)
        float tmax = score;
        #pragma unroll
        for (int off = 16; off > 0; off >>= 1)
            tmax = fmaxf(tmax, __shfl_xor(tmax, off, 32));
        const float m_new = fmaxf(m_run, tmax);
        const float scale = __expf(m_run - m_new);
        const float w     = __expf(score - m_new);
        float wsum = w;
        #pragma unroll
        for (int off = 16; off > 0; off >>= 1)
            wsum += __shfl_xor(wsum, off, 32);
        l_run = l_run * scale + 0.5f * wsum;
        o *= scale;
        m_run = m_new;

        // --- accumulate V: lane l owns d=4l..4l+3; row loads fully coalesced
        const float* vrow = vbase + (size_t)s0 * rs + 4 * lane;
        #pragma unroll
        for (int j = 0; j < 16; ++j) {
            const float wj = __shfl(w, j, 32);
            const v4f   v  = __builtin_nontemporal_load((const v4f*)(vrow + (size_t)j * rs));
            o += wj * v;
        }
    }

    // --- merge the 8 per-wave partials through LDS
    __shared__ float sm[8];
    __shared__ float sl[8];
    __shared__ float so[8 * HD];
    *(v4f*)(so + wave * HD + 4 * lane) = o;
    if (lane == 0) { sm[wave] = m_run; sl[wave] = l_run; }
    __syncthreads();

    if (wave == 0) {
        float M = sm[0];
        #pragma unroll
        for (int wv = 1; wv < 8; ++wv) M = fmaxf(M, sm[wv]);
        float ltot = 0.f;
        v4f   oacc = {0.f, 0.f, 0.f, 0.f};
        #pragma unroll
        for (int wv = 0; wv < 8; ++wv) {
            const float f = __expf(sm[wv] - M);
            ltot += f * sl[wv];
            oacc += f * (*(const v4f*)(so + wv * HD + 4 * lane));
        }
        const float inv = 1.0f / ltot;
        *(v4f*)(out + (size_t)bh * HD + 4 * lane) = oacc * inv;  // bh*128 = b*4096+h*128
    }
}

// ---------------------------------------------------------------------------
// Launch: inputs in setup_inputs() order:
//   0: hidden_states_q [8,1536] f32   1: key   [8,4096,32,128] f32
//   2: value [8,4096,32,128] f32      3: w_q   [4096,1536] f32
//   4: b_q   [4096] f32
// Output: [8, 4096] f32. Workspace: q intermediate (8*4096 floats = 128 KB).
// ---------------------------------------------------------------------------
extern "C" void kernel_launch(void* const* d_in, const int* in_sizes, int n_in,
                              void* d_out, int out_size, void* d_ws, size_t ws_size,
                              hipStream_t stream) {
    const float* hs  = (const float*)d_in[0];
    const float* key = (const float*)d_in[1];
    const float* val = (const float*)d_in[2];
    const float* wq  = (const float*)d_in[3];
    const float* bq  = (const float*)d_in[4];
    float* out  = (float*)d_out;
    float* q_ws = (float*)d_ws;                 // 8*4096 floats

    (void)in_sizes; (void)n_in; (void)out_size; (void)ws_size;

    qproj_wmma_kernel<<<NCOL / 16, 32, 0, stream>>>(hs, wq, bq, q_ws);
    attn_flash_wmma_kernel<<<BSZ * NH, 256, 0, stream>>>(q_ws, key, val, out);
}